// ForwardBackwardImputer_17454747091127
// MI455X (gfx1250) — compile-verified
//
#include <hip/hip_runtime.h>

// Forward-fill imputer for x[B=256, L=2048, D=128] float32.
// Reference reduces to: out[b,l,:] = x[b, idx_fwd[b,l], :],
//   idx_fwd = running max over l of (row-all-|x|<=1e-6 ? 0 : l).
// Memory-bound: ~512MB total -> ~22us at 23.3 TB/s. No matmul FLOPs, so no WMMA;
// CDNA5 paths: wave32 ballot, nontemporal (TH=NT) stores, global_prefetch_b8,
// LDS block scan. Phase 1 is software-pipelined 4 deep so each wave keeps
// 4x512B loads in flight (bandwidth roofline needs ~16KB/SIMD in flight).

typedef float v4f __attribute__((ext_vector_type(4)));

namespace {
constexpr int   kL        = 2048;
constexpr int   kD        = 128;
constexpr int   kVPerRow  = kD / 4;           // 32 float4 per row (one per lane)
constexpr int   kThreads  = 512;              // 16 waves (wave32)
constexpr int   kWaves    = kThreads / 32;    // 16
constexpr int   kRowsPerW = kL / kWaves;      // 128 rows per wave
constexpr int   kUnroll   = 4;                // loads in flight per wave
constexpr int   kPerT     = kL / kThreads;    // 4 scan elements per thread
constexpr float kAtol     = 1e-6f;            // isclose(x, 0) -> |x| <= atol
}

__global__ __launch_bounds__(kThreads)
void ForwardBackwardImputer_ffill_kernel(const float* __restrict__ x,
                                         float* __restrict__ out) {
    __shared__ int s_m[kL];        // idx values, then scanned in place
    __shared__ int s_tot[kThreads];

    const int b  = blockIdx.x;
    const int t  = threadIdx.x;
    const int w  = t >> 5;         // wave id within block
    const int ln = t & 31;         // lane id (wave32)

    const v4f* __restrict__ xv = (const v4f*)x;
    v4f* __restrict__ ov       = (v4f*)out;
    const size_t rb = (size_t)b * kL;   // row index base for this batch

    // ---- Phase 1: per-row missing mask; write-through rows already resolved ----
    // Wave w owns rows l = w + k*16, k = 0..127, processed in batches of 4 so
    // 4 global_load_b128 are outstanding while earlier rows are balloted/stored.
    for (int k = 0; k < kRowsPerW; k += kUnroll) {
        v4f v[kUnroll];
        #pragma unroll
        for (int u = 0; u < kUnroll; ++u) {
            const int l = w + (k + u) * kWaves;
            v[u] = xv[(rb + (size_t)l) * kVPerRow + ln];
        }
        // Unconditional stream-ahead prefetch of the next batch's first row;
        // wave-uniform clamped index avoids any exec-mask manipulation.
        {
            const int pk = (k + kUnroll < kRowsPerW) ? (k + kUnroll) : k;
            const int pl = w + pk * kWaves;
            __builtin_prefetch((const void*)&xv[(rb + (size_t)pl) * kVPerRow + ln], 0, 3);
        }
        #pragma unroll
        for (int u = 0; u < kUnroll; ++u) {
            const int l = w + (k + u) * kWaves;
            const size_t vidx = (rb + (size_t)l) * kVPerRow + ln;
            const bool ok = (__builtin_fabsf(v[u].x) <= kAtol) &&
                            (__builtin_fabsf(v[u].y) <= kAtol) &&
                            (__builtin_fabsf(v[u].z) <= kAtol) &&
                            (__builtin_fabsf(v[u].w) <= kAtol);
            const unsigned bal   = __builtin_amdgcn_ballot_w32(ok);
            const bool   missing = (bal == 0xffffffffu);   // all 128 features ~0
            // All lanes write the same wave-uniform value to the same LDS
            // address: deterministic (one writer wins, all values identical),
            // and avoids a per-row exec-mask toggle for "lane 0 only".
            s_m[l] = missing ? 0 : l;
            // Valid rows copy straight through. l==0 also writes when missing:
            // its gather source is row 0 itself (zeros), so v is already right.
            if (!missing || l == 0) {
                __builtin_nontemporal_store(v[u], &ov[vidx]);  // never re-read
            }
        }
    }
    __syncthreads();

    // ---- Phase 2a: per-thread serial max-scan of 4 consecutive elements ----
    const int base = t * kPerT;
    int run = s_m[base];
    #pragma unroll
    for (int i = 1; i < kPerT; ++i) {
        run = max(run, s_m[base + i]);
        s_m[base + i] = run;
    }
    s_tot[t] = run;
    __syncthreads();

    // ---- Phase 2b: Hillis-Steele inclusive max-scan over 512 thread totals ----
    for (int off = 1; off < kThreads; off <<= 1) {
        int v = s_tot[t];
        if (t >= off) v = max(v, s_tot[t - off]);
        __syncthreads();
        s_tot[t] = v;
        __syncthreads();
    }

    // ---- Phase 2c: apply exclusive prefix (identity 0 is safe: indices >= 0) ----
    const int prefix = (t > 0) ? s_tot[t - 1] : 0;
    #pragma unroll
    for (int i = 0; i < kPerT; ++i) {
        s_m[base + i] = max(s_m[base + i], prefix);
    }
    __syncthreads();

    // ---- Phase 3: gather only unresolved rows (src != l; ~10% of rows) ----
    for (int l = w; l < kL; l += kWaves) {
        const int src = s_m[l];            // uniform across the wave
        if (src != l) {                    // wave-uniform branch
            v4f v = xv[(rb + (size_t)src) * kVPerRow + ln];  // likely L2 hit
            __builtin_nontemporal_store(v, &ov[(rb + (size_t)l) * kVPerRow + ln]);
        }
    }
}

extern "C" void kernel_launch(void* const* d_in, const int* in_sizes, int n_in,
                              void* d_out, int out_size, void* d_ws, size_t ws_size,
                              hipStream_t stream) {
    (void)n_in; (void)out_size; (void)d_ws; (void)ws_size;
    const float* x = (const float*)d_in[0];
    float* out     = (float*)d_out;
    const int B    = in_sizes[0] / (kL * kD);   // 256
    ForwardBackwardImputer_ffill_kernel<<<dim3(B), dim3(kThreads), 0, stream>>>(x, out);
}